// SSM_80195629351494
// MI455X (gfx1250) — compile-verified
//
#include <hip/hip_runtime.h>
#include <hip/hip_bf16.h>
#include <math.h>

// ---------------------------------------------------------------------------
// Mamba-style selective-scan block on MI455X (gfx1250, wave32, WMMA).
// All GEMMs are X @ W^T in fp32 via V_WMMA_F32_16X16X4_F32.
// Big GEMMs use 64x64 register-blocked macro-tiles (16 WMMA tiles / wave):
// 8 b64 loads feed 16 WMMAs per K-step (0.5 loads/WMMA vs 2.0 unblocked).
// ---------------------------------------------------------------------------

typedef __attribute__((ext_vector_type(2))) float v2f;
typedef __attribute__((ext_vector_type(8))) float v8f;

#define B_    2
#define L_    2048
#define H_    1024
#define DIN_  2048
#define KC_   4
#define NS_   16
#define R_    64
#define M_    (B_ * L_)        // 4096 rows (b*l flattened)
#define XDBL_ (R_ + 2 * NS_)   // 96

__device__ __forceinline__ float silu_f(float x) {
    return x * (1.0f / (1.0f + __expf(-x)));
}

// ---------------------------------------------------------------------------
// 64x64 macro-tile GEMM: C[m,n] = sum_k A[m,k] * W[n,k]  (A @ W^T), fp32.
// One wave per 64x64 output block = 4x4 grid of 16x16 WMMA tiles.
//
// Fragment layouts (CDNA5 ISA 7.12.2, 32-bit 16x4 A / 4x16 B / 16x16 C):
//   lane = threadIdx.x, half = lane>>4, r = lane&15
//   A frag: lanes 0-15 -> (M=r, K=k0+0/1), lanes 16-31 -> (M=r, K=k0+2/3)
//   B frag mirrors A with N=r (since B[k][n] = W[n][k])
//   C VGPR v: row = m0 + v + 8*half, col = n0 + r
//
// mode 0: store acc ; mode 1: store softplus(acc + bias[col])
// Requires: M%64==0, N%64==0, K%64==0.
// ---------------------------------------------------------------------------
__global__ void __launch_bounds__(32)
wmma_gemm_xt_64(const float* __restrict__ A, int lda,
                const float* __restrict__ W, int ldw,
                float* __restrict__ C, int ldc,
                int K, int mode, const float* __restrict__ bias)
{
    const int n0 = blockIdx.x * 64;
    const int m0 = blockIdx.y * 64;
    const int lane = threadIdx.x;
    const int half = lane >> 4;
    const int r    = lane & 15;

    const float* __restrict__ arow[4];
    const float* __restrict__ brow[4];
    #pragma unroll
    for (int i = 0; i < 4; ++i) {
        arow[i] = A + (size_t)(m0 + i * 16 + r) * lda + 2 * half;
        brow[i] = W + (size_t)(n0 + i * 16 + r) * ldw + 2 * half;
    }

    v8f acc[4][4];
    #pragma unroll
    for (int i = 0; i < 4; ++i)
        #pragma unroll
        for (int j = 0; j < 4; ++j)
            acc[i][j] = (v8f){};

    for (int k0 = 0; k0 < K; k0 += 64) {
        #pragma unroll
        for (int i = 0; i < 4; ++i) {
            __builtin_prefetch(arow[i] + k0 + 64, 0, 1);   // global_prefetch_b8
            __builtin_prefetch(brow[i] + k0 + 64, 0, 1);
        }
        #pragma unroll 4
        for (int kk = 0; kk < 64; kk += 4) {
            v2f a[4], b[4];
            #pragma unroll
            for (int i = 0; i < 4; ++i) a[i] = *(const v2f*)(arow[i] + k0 + kk);
            #pragma unroll
            for (int j = 0; j < 4; ++j) b[j] = *(const v2f*)(brow[j] + k0 + kk);
            #pragma unroll
            for (int i = 0; i < 4; ++i)
                #pragma unroll
                for (int j = 0; j < 4; ++j)
                    acc[i][j] = __builtin_amdgcn_wmma_f32_16x16x4_f32(
                        false, a[i], false, b[j], (short)0, acc[i][j],
                        false, false);
        }
    }

    #pragma unroll
    for (int j = 0; j < 4; ++j) {
        const int col = n0 + j * 16 + r;
        const float bval = (mode == 1) ? bias[col] : 0.0f;
        #pragma unroll
        for (int i = 0; i < 4; ++i) {
            #pragma unroll
            for (int v = 0; v < 8; ++v) {
                const int row = m0 + i * 16 + v + 8 * half;
                float x = acc[i][j][v];
                if (mode == 1) {
                    x += bval;
                    x = (x > 20.0f) ? x : log1pf(__expf(x));   // softplus
                }
                C[(size_t)row * ldc + col] = x;
            }
        }
    }
}

// ---------------------------------------------------------------------------
// Single-tile GEMM (16x16 per wave) for narrow N (x_dbl: N = 96).
// ---------------------------------------------------------------------------
__global__ void __launch_bounds__(32)
wmma_gemm_xt_16(const float* __restrict__ A, int lda,
                const float* __restrict__ W, int ldw,
                float* __restrict__ C, int ldc, int K)
{
    const int tile_n = blockIdx.x;
    const int tile_m = blockIdx.y;
    const int lane = threadIdx.x;
    const int half = lane >> 4;
    const int r    = lane & 15;

    const float* __restrict__ arow = A + (size_t)(tile_m * 16 + r) * lda + 2 * half;
    const float* __restrict__ brow = W + (size_t)(tile_n * 16 + r) * ldw + 2 * half;

    v8f acc = {};
    for (int k0 = 0; k0 < K; k0 += 64) {
        __builtin_prefetch(arow + k0 + 64, 0, 1);
        __builtin_prefetch(brow + k0 + 64, 0, 1);
        #pragma unroll
        for (int kk = 0; kk < 64; kk += 4) {
            v2f a = *(const v2f*)(arow + k0 + kk);
            v2f b = *(const v2f*)(brow + k0 + kk);
            acc = __builtin_amdgcn_wmma_f32_16x16x4_f32(
                      false, a, false, b, (short)0, acc, false, false);
        }
    }

    const int col = tile_n * 16 + r;
    #pragma unroll
    for (int v = 0; v < 8; ++v) {
        const int row = tile_m * 16 + v + 8 * half;
        C[(size_t)row * ldc + col] = acc[v];
    }
}

// ---------------------------------------------------------------------------
// Depthwise causal conv1d (K=4) over L, per channel, + SiLU.
// hs is the first DIN_ columns of hs_res [M_, 2*DIN_].
// ---------------------------------------------------------------------------
__global__ void conv_silu_kernel(const float* __restrict__ hs_res,
                                 const float* __restrict__ wconv,
                                 float* __restrict__ hsc)
{
    const int idx = blockIdx.x * blockDim.x + threadIdx.x;  // [0, M_*DIN_)
    const int d  = idx & (DIN_ - 1);
    const int bl = idx >> 11;          // DIN_ = 2^11
    const int l  = bl & (L_ - 1);
    const int b0 = bl - l;             // b * L_

    const float* __restrict__ wr = wconv + d * KC_;
    float s = 0.0f;
    #pragma unroll
    for (int j = 0; j < KC_; ++j) {
        const int ls = l + j - (KC_ - 1);
        if (ls >= 0)
            s += wr[j] * hs_res[(size_t)(b0 + ls) * (2 * DIN_) + d];
    }
    hsc[idx] = silu_f(s);
}

// ---------------------------------------------------------------------------
// Selective scan: 16 lanes (one per state n) per (b,d) channel.
//   z_n <- exp(delta*A[d,n]) * z_n + delta * x * B_n ;  y = sum_n z_n * C_n
// Cross-lane sum over the 16-lane group via __shfl_xor (wave32).
// ---------------------------------------------------------------------------
__global__ void scan_kernel(const float* __restrict__ hsc,
                            const float* __restrict__ delta,
                            const float* __restrict__ xdbl,
                            const float* __restrict__ A_log,
                            const float* __restrict__ Dp,
                            float* __restrict__ y)
{
    const int g  = blockIdx.x * blockDim.x + threadIdx.x;   // [0, B_*DIN_*NS_)
    const int n  = g & (NS_ - 1);
    const int ch = g >> 4;
    const int d  = ch & (DIN_ - 1);
    const int b  = ch >> 11;

    const float Ad = -__expf(A_log[d * NS_ + n]);
    const float Dd = Dp[d];
    const size_t base0 = (size_t)b * L_;

    float z = 0.0f;
    for (int l = 0; l < L_; ++l) {
        const size_t rowm = base0 + l;
        const float dv = delta[rowm * DIN_ + d];
        const float xv = hsc[rowm * DIN_ + d];
        const float Bv = xdbl[rowm * XDBL_ + R_ + n];
        const float Cv = xdbl[rowm * XDBL_ + R_ + NS_ + n];

        const float dA = __expf(dv * Ad);
        z = fmaf(dA, z, dv * xv * Bv);

        float p = z * Cv;
        p += __shfl_xor(p, 1);
        p += __shfl_xor(p, 2);
        p += __shfl_xor(p, 4);
        p += __shfl_xor(p, 8);
        if (n == 0)
            y[rowm * DIN_ + d] = fmaf(xv, Dd, p);
    }
}

// ---------------------------------------------------------------------------
// Gate: y *= silu(res), res = hs_res[:, DIN_:]
// ---------------------------------------------------------------------------
__global__ void gate_kernel(float* __restrict__ y,
                            const float* __restrict__ hs_res)
{
    const int idx = blockIdx.x * blockDim.x + threadIdx.x;  // [0, M_*DIN_)
    const int d  = idx & (DIN_ - 1);
    const int bl = idx >> 11;
    const float rv = hs_res[(size_t)bl * (2 * DIN_) + DIN_ + d];
    y[idx] *= silu_f(rv);
}

// ---------------------------------------------------------------------------
// Host-side launcher
// ---------------------------------------------------------------------------
extern "C" void kernel_launch(void* const* d_in, const int* in_sizes, int n_in,
                              void* d_out, int out_size, void* d_ws, size_t ws_size,
                              hipStream_t stream)
{
    const float* hidden  = (const float*)d_in[0];  // (B,L,H)
    const float* w_in    = (const float*)d_in[1];  // (2*DIN, H)
    const float* w_conv  = (const float*)d_in[2];  // (DIN, 1, K)
    const float* w_xproj = (const float*)d_in[3];  // (96, DIN)
    const float* w_dt    = (const float*)d_in[4];  // (DIN, R)
    const float* b_dt    = (const float*)d_in[5];  // (DIN,)
    const float* A_log   = (const float*)d_in[6];  // (DIN, N)
    const float* Dp      = (const float*)d_in[7];  // (DIN,)
    const float* w_out   = (const float*)d_in[8];  // (H, DIN)
    float* out = (float*)d_out;                    // (B,L,H)

    float* ws      = (float*)d_ws;
    float* hs_res  = ws;                                     // M_ * 2*DIN_
    float* hsc     = hs_res + (size_t)M_ * 2 * DIN_;         // M_ * DIN_
    float* xdbl    = hsc    + (size_t)M_ * DIN_;             // M_ * 96
    float* deltab  = xdbl   + (size_t)M_ * XDBL_;            // M_ * DIN_
    float* ybuf    = deltab + (size_t)M_ * DIN_;             // M_ * DIN_

    const dim3 wb(32);

    // 1) hs_res = hidden @ w_in^T   [4096 x 4096], K = 1024
    wmma_gemm_xt_64<<<dim3(2 * DIN_ / 64, M_ / 64), wb, 0, stream>>>(
        hidden, H_, w_in, H_, hs_res, 2 * DIN_, H_, 0, nullptr);

    // 2) depthwise causal conv + SiLU -> hsc
    conv_silu_kernel<<<(M_ * DIN_) / 256, 256, 0, stream>>>(hs_res, w_conv, hsc);

    // 3) x_dbl = hsc @ w_xproj^T    [4096 x 96], K = 2048
    wmma_gemm_xt_16<<<dim3(XDBL_ / 16, M_ / 16), wb, 0, stream>>>(
        hsc, DIN_, w_xproj, DIN_, xdbl, XDBL_, DIN_);

    // 4) delta = softplus(dt @ w_dt^T + b_dt)   [4096 x 2048], K = 64
    //    dt = x_dbl[:, :64] -> lda = 96
    wmma_gemm_xt_64<<<dim3(DIN_ / 64, M_ / 64), wb, 0, stream>>>(
        xdbl, XDBL_, w_dt, R_, deltab, DIN_, R_, 1, b_dt);

    // 5) selective scan -> ybuf (includes + x*D)
    scan_kernel<<<(B_ * DIN_ * NS_) / 256, 256, 0, stream>>>(
        hsc, deltab, xdbl, A_log, Dp, ybuf);

    // 6) ybuf *= silu(res)
    gate_kernel<<<(M_ * DIN_) / 256, 256, 0, stream>>>(ybuf, hs_res);

    // 7) out = ybuf @ w_out^T       [4096 x 1024], K = 2048
    wmma_gemm_xt_64<<<dim3(H_ / 64, M_ / 64), wb, 0, stream>>>(
        ybuf, DIN_, w_out, DIN_, out, H_, DIN_, 0, nullptr);
}